// SAGE_23940147708109
// MI455X (gfx1250) — compile-verified
//
#include <hip/hip_runtime.h>
#include <hip/hip_bf16.h>

typedef __attribute__((ext_vector_type(2))) float v2f;
typedef __attribute__((ext_vector_type(8))) float v8f;

#define FEATS 128
#define LDS_STRIDE 132   // 132 % 64 == 4 -> conflict-free strided A reads

// ---------------------------------------------------------------- zero fill
__global__ void sage_zero_f4(float4* __restrict__ p, long long n4) {
    long long i = (long long)blockIdx.x * blockDim.x + threadIdx.x;
    if (i < n4) p[i] = make_float4(0.f, 0.f, 0.f, 0.f);
}

// ---------------------------------------------------------------- degree
__global__ void sage_degree(const int* __restrict__ dst, float* __restrict__ deg, int E) {
    int e = blockIdx.x * blockDim.x + threadIdx.x;
    if (e < E) atomicAdd(&deg[dst[e]], 1.0f);
}

__global__ void sage_invdeg(float* __restrict__ deg, int M) {
    int i = blockIdx.x * blockDim.x + threadIdx.x;
    if (i < M) deg[i] = 1.0f / fmaxf(deg[i], 1.0f);
}

// ---------------------------------------------------------------- scatter-sum
// One wave per edge: lane g handles float4 group g of the 128 features.
__global__ __launch_bounds__(256)
void sage_scatter(const float* __restrict__ h, const int* __restrict__ src,
                  const int* __restrict__ dst, float* __restrict__ agg, int E) {
    long long tid = (long long)blockIdx.x * blockDim.x + threadIdx.x;
    int e = (int)(tid >> 5);
    int g = (int)(tid & 31);
    if (e >= E) return;
    int s = src[e];
    int d = dst[e];
    const float4 v = ((const float4*)(h + (size_t)s * FEATS))[g];
    float* out = agg + (size_t)d * FEATS + g * 4;
    atomicAdd(out + 0, v.x);
    atomicAdd(out + 1, v.y);
    atomicAdd(out + 2, v.z);
    atomicAdd(out + 3, v.w);
}

// ---------------------------------------------------------------- fused dual GEMM
// out[i][j] = act( sum_k A[i][k]*Ws[k][j] + (agg[i][k]*invdeg[i])*Wn[k][j] + bias[j] )
// Block = 256 threads (8 waves); block owns 16 rows; wave w owns cols [16w,16w+16).
__global__ __launch_bounds__(256)
void sage_gemm(const float* __restrict__ A, const float* __restrict__ Agg,
               const float* __restrict__ invdeg,
               const float* __restrict__ Wself, const float* __restrict__ Wneigh,
               const float* __restrict__ bias, float* __restrict__ Out,
               int M, int N, int relu) {
    __shared__ float As[16 * LDS_STRIDE];
    __shared__ float An[16 * LDS_STRIDE];

    const int m0 = blockIdx.x * 16;
    const int t  = threadIdx.x;

    // cooperative staged load of the 16x128 self and neighbor tiles (coalesced)
    for (int i = t; i < 16 * FEATS; i += 256) {
        int row  = i >> 7;
        int colk = i & (FEATS - 1);
        int gr   = m0 + row;
        int grc  = gr < M ? gr : (M - 1);
        size_t gi = (size_t)grc * FEATS + colk;
        As[row * LDS_STRIDE + colk] = A[gi];
        An[row * LDS_STRIDE + colk] = Agg[gi] * invdeg[grc];
    }
    __syncthreads();

    const int wave = t >> 5;
    const int lane = t & 31;
    const int half = lane >> 4;   // 0: K even pair, 1: K odd pair (per ISA 32-bit layout)
    const int r    = lane & 15;   // A row within tile / B col within tile
    const int n0   = wave * 16;
    if (n0 >= N) return;
    const int colg = n0 + r;

    v8f c = {};
    // self-term: K = 0..127 over h tile
    for (int kk = 0; kk < FEATS; kk += 4) {
        int ka = kk + half * 2;
        v2f a, b;
        a.x = As[r * LDS_STRIDE + ka];
        a.y = As[r * LDS_STRIDE + ka + 1];
        b.x = Wself[ka * N + colg];
        b.y = Wself[(ka + 1) * N + colg];
        c = __builtin_amdgcn_wmma_f32_16x16x4_f32(false, a, false, b,
                                                  (short)0, c, false, false);
    }
    // neighbor-term: K = 0..127 over (agg*invdeg) tile, same accumulator
    for (int kk = 0; kk < FEATS; kk += 4) {
        int ka = kk + half * 2;
        v2f a, b;
        a.x = An[r * LDS_STRIDE + ka];
        a.y = An[r * LDS_STRIDE + ka + 1];
        b.x = Wneigh[ka * N + colg];
        b.y = Wneigh[(ka + 1) * N + colg];
        c = __builtin_amdgcn_wmma_f32_16x16x4_f32(false, a, false, b,
                                                  (short)0, c, false, false);
    }

    const float bb = bias[colg];
    // C/D layout: VGPR i holds M = i (lanes 0-15) / M = 8+i (lanes 16-31), N = r
    #pragma unroll
    for (int i = 0; i < 8; ++i) {
        int row = m0 + half * 8 + i;
        if (row >= M) continue;
        float v = c[i] + bb;
        if (relu) v = fmaxf(v, 0.0f);
        Out[(size_t)row * N + colg] = v;
    }
}

// ---------------------------------------------------------------- launcher
extern "C" void kernel_launch(void* const* d_in, const int* in_sizes, int n_in,
                              void* d_out, int out_size, void* d_ws, size_t ws_size,
                              hipStream_t stream) {
    const float* x   = (const float*)d_in[0];
    const int*   src = (const int*)  d_in[1];
    const int*   dst = (const int*)  d_in[2];
    const float* Ws0 = (const float*)d_in[3];
    const float* Wn0 = (const float*)d_in[4];
    const float* b0  = (const float*)d_in[5];
    const float* Ws1 = (const float*)d_in[6];
    const float* Wn1 = (const float*)d_in[7];
    const float* b1  = (const float*)d_in[8];
    const float* Ws2 = (const float*)d_in[9];
    const float* Wn2 = (const float*)d_in[10];
    const float* b2  = (const float*)d_in[11];

    const int M = in_sizes[0] / FEATS;   // 100000
    const int E = in_sizes[1];           // 1600000
    const int NC = out_size / M;         // 64

    char* ws = (char*)d_ws;
    float* hbuf = (float*)ws;                                 // M*128 f32
    float* agg  = (float*)(ws + (size_t)M * FEATS * 4);       // M*128 f32
    float* deg  = (float*)(ws + (size_t)M * FEATS * 4 * 2);   // M f32

    const int TB = 256;
    const long long aggN4 = (long long)M * FEATS / 4;
    const long long degN4 = (M + 3) / 4;
    dim3 zeroAggGrid((unsigned)((aggN4 + TB - 1) / TB));
    dim3 zeroDegGrid((unsigned)((degN4 + TB - 1) / TB));
    dim3 edgeGrid((E + TB - 1) / TB);
    dim3 nodeGrid((M + TB - 1) / TB);
    dim3 scatGrid((unsigned)(((long long)E * 32 + TB - 1) / TB));
    dim3 gemmGrid((M + 15) / 16);

    // degree -> 1/max(deg,1), once
    sage_zero_f4<<<zeroDegGrid, TB, 0, stream>>>((float4*)deg, degN4);
    sage_degree<<<edgeGrid, TB, 0, stream>>>(dst, deg, E);
    sage_invdeg<<<nodeGrid, TB, 0, stream>>>(deg, M);

    // layer 0: x -> hbuf (relu)
    sage_zero_f4<<<zeroAggGrid, TB, 0, stream>>>((float4*)agg, aggN4);
    sage_scatter<<<scatGrid, TB, 0, stream>>>(x, src, dst, agg, E);
    sage_gemm<<<gemmGrid, TB, 0, stream>>>(x, agg, deg, Ws0, Wn0, b0, hbuf,
                                           M, FEATS, 1);

    // layer 1: hbuf -> hbuf in place (relu); safe: blocks read only rows they write
    sage_zero_f4<<<zeroAggGrid, TB, 0, stream>>>((float4*)agg, aggN4);
    sage_scatter<<<scatGrid, TB, 0, stream>>>(hbuf, src, dst, agg, E);
    sage_gemm<<<gemmGrid, TB, 0, stream>>>(hbuf, agg, deg, Ws1, Wn1, b1, hbuf,
                                           M, FEATS, 1);

    // layer 2: hbuf -> d_out (no relu), N = 64
    sage_zero_f4<<<zeroAggGrid, TB, 0, stream>>>((float4*)agg, aggN4);
    sage_scatter<<<scatGrid, TB, 0, stream>>>(hbuf, src, dst, agg, E);
    sage_gemm<<<gemmGrid, TB, 0, stream>>>(hbuf, agg, deg, Ws2, Wn2, b2,
                                           (float*)d_out, M, NC, 0);
}